// TAG_FC3_L3_70574902608028
// MI455X (gfx1250) — compile-verified
//
#include <hip/hip_runtime.h>
#include <hip/hip_bf16.h>

#define NFEAT 16
#define HID   96
#define OUTF  4
#define KHOP  3

typedef __attribute__((ext_vector_type(2))) float v2f;
typedef __attribute__((ext_vector_type(8))) float v8f;

// ---------------- utility ----------------
__global__ void zero_kernel(float* __restrict__ p, int n) {
    int i = blockIdx.x * blockDim.x + threadIdx.x;
    if (i < n) p[i] = 0.0f;
}

// deg[col[e]] += ew[e]
__global__ void degree_kernel(const long long* __restrict__ col,
                              const float* __restrict__ ew,
                              float* __restrict__ deg, int nedges) {
    int e = blockIdx.x * blockDim.x + threadIdx.x;
    if (e < nedges) atomicAdd(&deg[(int)col[e]], ew[e]);
}

// deg -> rsqrt(deg) in place (0 where deg<=0)
__global__ void dinv_kernel(float* __restrict__ deg, int n) {
    int i = blockIdx.x * blockDim.x + threadIdx.x;
    if (i < n) {
        float d = deg[i];
        deg[i] = (d > 0.0f) ? rsqrtf(d) : 0.0f;
    }
}

// norm[e] = dinv[row]*ew*dinv[col]
__global__ void norm_kernel(const long long* __restrict__ row,
                            const long long* __restrict__ col,
                            const float* __restrict__ ew,
                            const float* __restrict__ dinv,
                            float* __restrict__ nrm, int nedges) {
    int e = blockIdx.x * blockDim.x + threadIdx.x;
    if (e < nedges)
        nrm[e] = dinv[(int)row[e]] * ew[e] * dinv[(int)col[e]];
}

// out[col[e], :] += norm[e] * H[row[e], :]   (one wave per edge, lane-strided features)
__global__ void spmm_kernel(const float* __restrict__ H,
                            const long long* __restrict__ row,
                            const long long* __restrict__ col,
                            const float* __restrict__ nrm,
                            float* __restrict__ out, int nedges, int f) {
    int wid  = (blockIdx.x * blockDim.x + threadIdx.x) >> 5;
    int lane = threadIdx.x & 31;
    if (wid >= nedges) return;
    int r = (int)row[wid];
    int c = (int)col[wid];
    float w = nrm[wid];
    const float* src = H + (size_t)r * f;
    float* dst = out + (size_t)c * f;
    for (int j = lane; j < f; j += 32)
        atomicAdd(&dst[j], w * src[j]);
}

// ---------------- WMMA GEMM (fout = 96, fin = 16 or 96, fp32) ----------------
// Block = 128 threads (4 waves). Block stages W[FIN x 96] in LDS once (padded
// stride 104 -> lanes 0-15 / 16-31 hit disjoint bank halves). Each wave owns a
// contiguous 16-row strip of A, copies it coalesced into its private LDS slab
// (padded stride FIN+1), then runs 6 column tiles with V_WMMA_F32_16X16X4_F32.
// A layout per ISA: lanes 0-15 = (M=lane, K=k+0/+1); lanes 16-31 = (M=lane-16, K=k+2/+3).
// D layout: vgpr r -> M=r (lanes<16) / M=8+r (lanes>=16), N = lane&15.
template <int FIN>
__global__ void gemm_wmma_lds(const float* __restrict__ A,
                              const float* __restrict__ W,
                              const float* __restrict__ bias,
                              float* __restrict__ C,
                              int n, int accum, int relu) {
    constexpr int FOUT = 96;
    constexpr int WS   = FOUT + 8;   // 104: rows k and k+2 differ by 208 dw = 16 banks
    constexpr int AS   = FIN + 1;    // odd stride -> rows spread across banks

    __shared__ float sW[FIN * WS];
    __shared__ float sA[4][16 * AS];

    const int tid  = threadIdx.x;
    const int wave = tid >> 5;
    const int lane = tid & 31;

    // cooperative, fully coalesced W fill (FIN*96 is a multiple of 128)
    for (int i = tid; i < FIN * FOUT; i += 128) {
        int r = i / FOUT, cc = i - r * FOUT;
        sW[r * WS + cc] = W[i];
    }
    __syncthreads();

    const int tm = blockIdx.x * 4 + wave;          // 16-row tile index
    if (tm * 16 >= n) return;                      // wave-uniform after the barrier

    // per-wave A strip: 16*FIN contiguous floats in global -> coalesced copy
    float* myA = &sA[wave][0];
    const float* gA = A + (size_t)tm * 16 * FIN;
    for (int i = lane; i < 16 * FIN; i += 32) {
        int r = i / FIN, cc = i - r * FIN;
        myA[r * AS + cc] = gA[i];
    }
    // wave-private LDS: compiler-inserted s_wait_dscnt orders these reads

    const int nlo = lane & 15;
    const int kk  = (lane >> 4) << 1;              // 0 (lanes 0-15) / 2 (lanes 16-31)

    v8f c[6];
#pragma unroll
    for (int t = 0; t < 6; ++t) c[t] = v8f{0.f,0.f,0.f,0.f,0.f,0.f,0.f,0.f};

    for (int k = 0; k < FIN; k += 4) {
        v2f a;
        a.x = myA[nlo * AS + k + kk];
        a.y = myA[nlo * AS + k + kk + 1];
#pragma unroll
        for (int t = 0; t < 6; ++t) {
            v2f b;
            b.x = sW[(k + kk)     * WS + t * 16 + nlo];
            b.y = sW[(k + kk + 1) * WS + t * 16 + nlo];
            c[t] = __builtin_amdgcn_wmma_f32_16x16x4_f32(
                       false, a, false, b, (short)0, c[t], false, false);
        }
    }

    const int rbase = (lane >> 4) << 3;            // rows 0-7 or 8-15 of the tile
#pragma unroll
    for (int t = 0; t < 6; ++t) {
        int ncol = t * 16 + nlo;
        float bv = bias ? bias[ncol] : 0.0f;
#pragma unroll
        for (int r = 0; r < 8; ++r) {
            int m = tm * 16 + rbase + r;
            size_t idx = (size_t)m * FOUT + ncol;
            float v = c[t][r] + bv;
            if (accum) v += C[idx];
            if (relu)  v = v > 0.0f ? v : 0.0f;
            C[idx] = v;
        }
    }
}

// final 96 -> 4 projection (tiny; plain VALU)
__global__ void fc_out_kernel(const float* __restrict__ H,
                              const float* __restrict__ W,
                              const float* __restrict__ b,
                              float* __restrict__ out, int n) {
    int i = blockIdx.x * blockDim.x + threadIdx.x;
    if (i >= n) return;
    float a0 = b[0], a1 = b[1], a2 = b[2], a3 = b[3];
    const float* h = H + (size_t)i * HID;
    for (int k = 0; k < HID; ++k) {
        float v = h[k];
        a0 += v * W[k * OUTF + 0];
        a1 += v * W[k * OUTF + 1];
        a2 += v * W[k * OUTF + 2];
        a3 += v * W[k * OUTF + 3];
    }
    out[(size_t)i * OUTF + 0] = a0;
    out[(size_t)i * OUTF + 1] = a1;
    out[(size_t)i * OUTF + 2] = a2;
    out[(size_t)i * OUTF + 3] = a3;
}

// ---------------- host side ----------------
static inline void launch_zero(float* p, int n, hipStream_t s) {
    zero_kernel<<<(n + 255) / 256, 256, 0, s>>>(p, n);
}

static inline void launch_gemm(const float* A, const float* W, const float* bias,
                               float* C, int n, int fin,
                               int accum, int relu, hipStream_t s) {
    int tiles  = (n + 15) / 16;
    int blocks = (tiles + 3) / 4;
    if (fin == NFEAT)
        gemm_wmma_lds<NFEAT><<<blocks, 128, 0, s>>>(A, W, bias, C, n, accum, relu);
    else
        gemm_wmma_lds<HID><<<blocks, 128, 0, s>>>(A, W, bias, C, n, accum, relu);
}

extern "C" void kernel_launch(void* const* d_in, const int* in_sizes, int n_in,
                              void* d_out, int out_size, void* d_ws, size_t ws_size,
                              hipStream_t stream) {
    const float*      x    = (const float*)d_in[0];
    const long long*  eidx = (const long long*)d_in[1];   // int64 [2, E]
    const float*      ew   = (const float*)d_in[2];
    const float*      W0   = (const float*)d_in[3];
    const float*      b0   = (const float*)d_in[4];
    const float*      W1   = (const float*)d_in[5];
    const float*      b1   = (const float*)d_in[6];
    const float*      W2   = (const float*)d_in[7];
    const float*      b2   = (const float*)d_in[8];
    const float*      fc1W = (const float*)d_in[9];
    const float*      fc1b = (const float*)d_in[10];
    const float*      fc2W = (const float*)d_in[11];
    const float*      fc2b = (const float*)d_in[12];
    const float*      fc4W = (const float*)d_in[13];
    const float*      fc4b = (const float*)d_in[14];
    const float*      fc5W = (const float*)d_in[15];
    const float*      fc5b = (const float*)d_in[16];
    float* out = (float*)d_out;

    const int N = in_sizes[0] / NFEAT;
    const int E = in_sizes[1] / 2;
    const long long* row = eidx;
    const long long* col = eidx + E;

    // workspace carve-up (floats)
    float* ws = (float*)d_ws;
    size_t off = 0;
    float* nrm = ws + off; off += (size_t)E;
    float* deg = ws + off; off += (size_t)N;
    off = (off + 63) & ~(size_t)63;
    float* bufA = ws + off; off += (size_t)N * HID;
    float* bufB = ws + off; off += (size_t)N * HID;
    float* bufC = ws + off; off += (size_t)N * HID;
    float* bufD = ws + off; off += (size_t)N * HID;

    // ---- gcn_norm ----
    launch_zero(deg, N, stream);
    degree_kernel<<<(E + 255) / 256, 256, 0, stream>>>(col, ew, deg, E);
    dinv_kernel<<<(N + 255) / 256, 256, 0, stream>>>(deg, N);
    norm_kernel<<<(E + 255) / 256, 256, 0, stream>>>(row, col, ew, deg, nrm, E);

    const int spmm_blocks = (int)(((long long)E * 32 + 255) / 256);

    // ---- one TAGConv layer: outbuf = sum_k (A^k in) @ W[k] + bias (+relu) ----
    auto tag_layer = [&](const float* in, int fin, const float* W, const float* bias,
                         float* outbuf, float* p1, float* p2, int relu) {
        launch_gemm(in, W, nullptr, outbuf, N, fin, /*accum=*/0, /*relu=*/0, stream);
        const float* hprev = in;
        float* dst = p1;
        for (int k = 1; k <= KHOP; ++k) {
            launch_zero(dst, N * fin, stream);
            spmm_kernel<<<spmm_blocks, 256, 0, stream>>>(hprev, row, col, nrm, dst, E, fin);
            int last = (k == KHOP);
            launch_gemm(dst, W + (size_t)k * fin * HID, last ? bias : nullptr,
                        outbuf, N, fin, /*accum=*/1, /*relu=*/last ? relu : 0, stream);
            hprev = dst;
            dst = (dst == p1) ? p2 : p1;
        }
    };

    tag_layer(x,    NFEAT, W0, b0, bufA, bufB, bufC, /*relu=*/1);
    tag_layer(bufA, HID,   W1, b1, bufD, bufB, bufC, /*relu=*/1);
    tag_layer(bufD, HID,   W2, b2, bufA, bufB, bufC, /*relu=*/0);

    // ---- FC stack: fc1, fc2, fc4, fc4 (all relu), then fc5 ----
    launch_gemm(bufA, fc1W, fc1b, bufD, N, HID, 0, 1, stream);
    launch_gemm(bufD, fc2W, fc2b, bufA, N, HID, 0, 1, stream);
    launch_gemm(bufA, fc4W, fc4b, bufD, N, HID, 0, 1, stream);
    launch_gemm(bufD, fc4W, fc4b, bufA, N, HID, 0, 1, stream);
    fc_out_kernel<<<(N + 255) / 256, 256, 0, stream>>>(bufA, fc5W, fc5b, out, N);
}